// GNN_31610959299128
// MI455X (gfx1250) — compile-verified
//
#include <hip/hip_runtime.h>

typedef float v2f __attribute__((ext_vector_type(2)));
typedef float v8f __attribute__((ext_vector_type(8)));

#define IN_CH  128
#define OUT_CH 64

// ---------------- degree kernels ----------------
__global__ void k_deg_init(float* __restrict__ deg, int n) {
    int i = blockIdx.x * blockDim.x + threadIdx.x;
    if (i < n) deg[i] = 1.0f;   // self-loop contributes 1 to every node's degree
}

__global__ void k_deg_count(const int* __restrict__ col, float* __restrict__ deg, int e) {
    int i = blockIdx.x * blockDim.x + threadIdx.x;
    if (i < e) atomicAdd(&deg[col[i]], 1.0f);
}

__global__ void k_deg_rsqrt(float* __restrict__ deg, int n) {
    int i = blockIdx.x * blockDim.x + threadIdx.x;
    if (i < n) deg[i] = rsqrtf(deg[i]);  // deg >= 1 always (self-loop)
}

// ---------------- h = x @ W via f32 WMMA ----------------
// Block = 256 threads = 8 waves. Wave w computes rows [row0, row0+16) x all 64 cols.
// Each wave: 4 accumulators (16x16 tiles), K-loop of 32 steps of V_WMMA_F32_16X16X4_F32.
// x is streamed with non-temporal loads (read-once, 51 MB) so L2 stays populated
// with h/out for the scatter phase.
__global__ void __launch_bounds__(256) k_gemm_wmma(const float* __restrict__ x,
                                                   const float* __restrict__ w,
                                                   float* __restrict__ h, int n) {
    const int wave = threadIdx.x >> 5;
    const int lane = threadIdx.x & 31;
    const int row0 = blockIdx.x * 128 + wave * 16;

    const int m     = lane & 15;          // A-matrix row within tile
    const int khalf = (lane >> 4) << 1;   // 0 or 2: which K pair this half-wave holds
    const int nn    = lane & 15;          // B/C/D column within 16-wide tile

    int arow = row0 + m;
    if (arow >= n) arow = n - 1;          // clamp: WMMA needs EXEC all-1s, guard stores instead
    const float* __restrict__ xr = x + (long)arow * IN_CH;

    v8f acc0 = {}; v8f acc1 = {}; v8f acc2 = {}; v8f acc3 = {};

    #pragma unroll 4
    for (int k = 0; k < IN_CH; k += 4) {
        v2f a = __builtin_nontemporal_load((const v2f*)(xr + k + khalf));

        const float* __restrict__ w0 = w + (long)(k + khalf) * OUT_CH;
        const float* __restrict__ w1 = w0 + OUT_CH;

        v2f b0; b0.x = w0[nn +  0]; b0.y = w1[nn +  0];
        v2f b1; b1.x = w0[nn + 16]; b1.y = w1[nn + 16];
        v2f b2; b2.x = w0[nn + 32]; b2.y = w1[nn + 32];
        v2f b3; b3.x = w0[nn + 48]; b3.y = w1[nn + 48];

        acc0 = __builtin_amdgcn_wmma_f32_16x16x4_f32(false, a, false, b0, (short)0, acc0, false, false);
        acc1 = __builtin_amdgcn_wmma_f32_16x16x4_f32(false, a, false, b1, (short)0, acc1, false, false);
        acc2 = __builtin_amdgcn_wmma_f32_16x16x4_f32(false, a, false, b2, (short)0, acc2, false, false);
        acc3 = __builtin_amdgcn_wmma_f32_16x16x4_f32(false, a, false, b3, (short)0, acc3, false, false);
    }

    // C/D layout: VGPR v, lane l -> M = v + 8*(l>>4), N = l&15
    const int mbase = (lane >> 4) * 8;
    #pragma unroll
    for (int v = 0; v < 8; ++v) {
        int r = row0 + mbase + v;
        if (r < n) {
            float* __restrict__ hp = h + (long)r * OUT_CH + nn;
            hp[ 0] = acc0[v];
            hp[16] = acc1[v];
            hp[32] = acc2[v];
            hp[48] = acc3[v];
        }
    }
}

// ---------------- out init with self-loop term (float4 vectorized) ----------------
__global__ void k_self_init(const float4* __restrict__ h4, const float* __restrict__ dinv,
                            float4* __restrict__ out4, int total4) {
    int i = blockIdx.x * blockDim.x + threadIdx.x;
    if (i < total4) {
        int node = i >> 4;            // 16 float4 per node (64 channels)
        float d = dinv[node];
        float s = d * d;
        float4 v = h4[i];
        v.x *= s; v.y *= s; v.z *= s; v.w *= s;
        out4[i] = v;
    }
}

// ---------------- edge scatter: one wave32 per edge, 2 channels per lane ----------------
__global__ void __launch_bounds__(256) k_scatter(const int* __restrict__ row, const int* __restrict__ col,
                                                 const float* __restrict__ h, const float* __restrict__ dinv,
                                                 float* __restrict__ out, int e) {
    int eid  = blockIdx.x * 8 + (threadIdx.x >> 5);   // wave id == edge id
    int lane = threadIdx.x & 31;
    if (eid >= e) return;                             // whole wave exits together

    int r = row[eid];                                 // wave-uniform loads: one cacheline req/wave
    int c = col[eid];
    float nrm = dinv[r] * dinv[c];

    float2 hv = *((const float2*)(h + (long)r * OUT_CH) + lane);  // coalesced b64 gather
    float* op = out + (long)c * OUT_CH + lane * 2;
    atomicAdd(op,     hv.x * nrm);
    atomicAdd(op + 1, hv.y * nrm);
}

// ---------------- bias + relu + log_softmax: one wave32 per node ----------------
__global__ void k_finalize(float* __restrict__ out, const float* __restrict__ bias, int n) {
    int node = (int)((blockIdx.x * blockDim.x + threadIdx.x) >> 5);
    int lane = threadIdx.x & 31;
    if (node >= n) return;   // whole wave exits together (blockDim multiple of 32)

    float* __restrict__ p = out + (long)node * OUT_CH;
    float v0 = p[lane]      + bias[lane];
    float v1 = p[lane + 32] + bias[lane + 32];
    v0 = fmaxf(v0, 0.0f);
    v1 = fmaxf(v1, 0.0f);

    float mx = fmaxf(v0, v1);
    #pragma unroll
    for (int off = 16; off >= 1; off >>= 1)
        mx = fmaxf(mx, __shfl_xor(mx, off, 32));

    float s = expf(v0 - mx) + expf(v1 - mx);
    #pragma unroll
    for (int off = 16; off >= 1; off >>= 1)
        s += __shfl_xor(s, off, 32);

    float lse = mx + logf(s);
    p[lane]      = v0 - lse;
    p[lane + 32] = v1 - lse;
}

extern "C" void kernel_launch(void* const* d_in, const int* in_sizes, int n_in,
                              void* d_out, int out_size, void* d_ws, size_t ws_size,
                              hipStream_t stream) {
    const float* x  = (const float*)d_in[0];
    const int*   ei = (const int*)  d_in[1];
    const float* W  = (const float*)d_in[2];
    const float* b  = (const float*)d_in[3];

    const int n = in_sizes[0] / IN_CH;   // 100000
    const int e = in_sizes[1] / 2;       // 1600000
    const int* row = ei;                 // edge_index[0] = src
    const int* col = ei + e;             // edge_index[1] = dst

    float* out  = (float*)d_out;
    float* h    = (float*)d_ws;                                        // n*64 f32 = 25.6 MB
    float* dinv = (float*)((char*)d_ws + (size_t)n * OUT_CH * sizeof(float));  // n f32

    // 1) degrees (in-degree over col, incl. self-loop), then rsqrt in place
    k_deg_init <<<(n + 255) / 256, 256, 0, stream>>>(dinv, n);
    k_deg_count<<<(e + 255) / 256, 256, 0, stream>>>(col, dinv, e);
    k_deg_rsqrt<<<(n + 255) / 256, 256, 0, stream>>>(dinv, n);

    // 2) h = x @ W  (f32 WMMA, 128 rows per block)
    k_gemm_wmma<<<(n + 127) / 128, 256, 0, stream>>>(x, W, h, n);

    // 3) out = self-loop contribution, then scatter-add edge messages
    const int total4 = n * OUT_CH / 4;
    k_self_init<<<(total4 + 255) / 256, 256, 0, stream>>>((const float4*)h, dinv,
                                                          (float4*)out, total4);

    k_scatter<<<(e + 7) / 8, 256, 0, stream>>>(row, col, h, dinv, out, e);

    // 4) bias + relu + log_softmax (one wave per node, 8 nodes per block)
    k_finalize<<<(n + 7) / 8, 256, 0, stream>>>(out, b, n);

    (void)n_in; (void)out_size; (void)ws_size;
}